// CapsDense_58789512347928
// MI455X (gfx1250) — compile-verified
//
#include <hip/hip_runtime.h>

typedef __attribute__((ext_vector_type(2))) float v2f;
typedef __attribute__((ext_vector_type(8))) float v8f;

#define B_SZ  128
#define JCAPS 2048
#define KLEN  8
#define UNITS 32
#define QLEN  16
#define CH    16            // j-chunks for the s GEMM (2048/16 = 128 j each)

// ------------------------------------------------------------------
// one-time transpose: xt[j,k,i] = x[i,j,k]   (coalesced writes)
__global__ __launch_bounds__(256) void k_transpose_x(const float* __restrict__ x,
                                                     float* __restrict__ xt) {
    int t  = blockIdx.x * blockDim.x + threadIdx.x;   // over 2M elements, xt order
    int i  = t & (B_SZ - 1);
    int jk = t >> 7;
    int j  = jk >> 3, k = jk & 7;
    xt[t] = x[((size_t)i * JCAPS + j) * KLEN + k];
}

__global__ __launch_bounds__(256) void k_zero(float* __restrict__ p, int n) {
    int t = blockIdx.x * blockDim.x + threadIdx.x;
    if (t < n) p[t] = 0.f;
}

// softmax over the 32 units of one row j per wave (wave32: lane == unit)
__global__ __launch_bounds__(256) void k_softmax(const float* __restrict__ b,
                                                 float* __restrict__ c) {
    int wave = (blockIdx.x * blockDim.x + threadIdx.x) >> 5;  // == j, 0..2047
    int lane = threadIdx.x & 31;
    float val = b[(size_t)wave * UNITS + lane];
    float m = val;
    for (int off = 16; off; off >>= 1) m = fmaxf(m, __shfl_xor(m, off, 32));
    float e = expf(val - m);
    float sum = e;
    for (int off = 16; off; off >>= 1) sum += __shfl_xor(sum, off, 32);
    c[(size_t)wave * UNITS + lane] = e / sum;
}

// ------------------------------------------------------------------
// s GEMM: for one unit p and one j-chunk, each wave owns a 16-batch tile:
//   C[16 x 16(q)] += A(x)[16 x 4] * B(c*W)[4 x 16]  over kflat = j*8+k
__global__ __launch_bounds__(256) void k_s_gemm(const float* __restrict__ x,
                                                const float* __restrict__ W,
                                                const float* __restrict__ c,
                                                float* __restrict__ pbuf) {
    const int p    = blockIdx.x & (UNITS - 1);
    const int ch   = blockIdx.x >> 5;
    const int wave = threadIdx.x >> 5;
    const int lane = threadIdx.x & 31;
    const int mbase = wave << 4;         // batch-row tile (8 waves cover 128)
    const int half  = lane >> 4;         // 0: lanes 0-15, 1: lanes 16-31
    const int l15   = lane & 15;

    v8f acc = {0.f, 0.f, 0.f, 0.f, 0.f, 0.f, 0.f, 0.f};
    const int j0 = ch * (JCAPS / CH);
    for (int jj = 0; jj < JCAPS / CH; ++jj) {
        const int j = j0 + jj;
        const float cjp = c[(size_t)j * UNITS + p];
#pragma unroll
        for (int kk = 0; kk < KLEN; kk += 4) {
            const int k0 = kk + (half << 1);          // A/B lane-half K split
            // A: row = mbase + l15, K = k0, k0+1 (contiguous in x)
            const float* xp = x + ((size_t)(mbase + l15) * JCAPS + j) * KLEN + k0;
            v2f a; a.x = xp[0]; a.y = xp[1];
            // B: N = q = l15, K = k0 (VGPR0), k0+1 (VGPR1); scale by c[j,p]
            const float* wp = W + (((size_t)j * UNITS + p) * KLEN + k0) * QLEN + l15;
            v2f bf; bf.x = cjp * wp[0]; bf.y = cjp * wp[QLEN];
            acc = __builtin_amdgcn_wmma_f32_16x16x4_f32(false, a, false, bf,
                                                        (short)0, acc, false, false);
        }
    }
    // store partial C tile: VGPR r -> (i = mbase + r + 8*half, q = l15)
    float* pb = pbuf + (size_t)ch * (B_SZ * UNITS * QLEN);
#pragma unroll
    for (int r = 0; r < 8; ++r) {
        const int i = mbase + r + (half << 3);
        pb[((size_t)i * UNITS + p) * QLEN + l15] = acc[r];
    }
}

// deterministic split-K reduction of the CH partial buffers
__global__ __launch_bounds__(256) void k_reduce_s(const float* __restrict__ pbuf,
                                                  float* __restrict__ s) {
    int t = blockIdx.x * blockDim.x + threadIdx.x;    // 65536
    float acc = 0.f;
#pragma unroll
    for (int chn = 0; chn < CH; ++chn)
        acc += pbuf[(size_t)chn * (B_SZ * UNITS * QLEN) + t];
    s[t] = acc;
}

// v = ||s|| * s / (1 + ||s||^2), one (i,p) row of 16 per thread
__global__ __launch_bounds__(256) void k_squash(const float* __restrict__ s,
                                                float* __restrict__ v) {
    int t = blockIdx.x * blockDim.x + threadIdx.x;    // 4096 rows
    const float* sp = s + (size_t)t * QLEN;
    float n2 = 0.f;
#pragma unroll
    for (int q = 0; q < QLEN; ++q) n2 += sp[q] * sp[q];
    const float scale = sqrtf(n2) / (1.f + n2);
    float* vp = v + (size_t)t * QLEN;
#pragma unroll
    for (int q = 0; q < QLEN; ++q) vp[q] = scale * sp[q];
}

// ------------------------------------------------------------------
// b update: per (j-pair, unit p) wave:
//   C[(j',k) x q] = Xjᵀ[16 x 128] * V[128 x 16]   (K = batch, 32 WMMA steps)
//   b[j,p] += Σ_{k,q} W[j,p,k,q] * C[(j,k),q]     (lane-local dot + shfl reduce)
__global__ __launch_bounds__(256) void k_b_update(const float* __restrict__ xt,
                                                  const float* __restrict__ vbuf,
                                                  const float* __restrict__ W,
                                                  float* __restrict__ b) {
    const int gw    = (blockIdx.x * blockDim.x + threadIdx.x) >> 5;  // global wave
    const int lane  = threadIdx.x & 31;
    const int p     = gw & (UNITS - 1);
    const int jpair = gw >> 5;                 // 0..1023
    const int j0    = jpair << 1;
    const int half  = lane >> 4;
    const int l15   = lane & 15;

    // A row for this lane: rowA = l15 -> (j' = rowA>>3, k = rowA&7)
    const int jA = j0 + (l15 >> 3);
    const int kA = l15 & 7;
    const float* xrow = xt + ((size_t)jA * KLEN + kA) * B_SZ;   // + i (contiguous)

    v8f acc = {0.f, 0.f, 0.f, 0.f, 0.f, 0.f, 0.f, 0.f};
    for (int ib = 0; ib < B_SZ; ib += 4) {
        const int i0 = ib + (half << 1);
        v2f a; a.x = xrow[i0]; a.y = xrow[i0 + 1];
        // B: K = i0 (VGPR0), i0+1 (VGPR1); N = q = l15
        const float* vp = vbuf + ((size_t)i0 * UNITS + p) * QLEN + l15;
        v2f bf; bf.x = vp[0]; bf.y = vp[UNITS * QLEN];
        acc = __builtin_amdgcn_wmma_f32_16x16x4_f32(false, a, false, bf,
                                                    (short)0, acc, false, false);
    }
    // C: lane half -> j = j0+half, VGPR r -> k = r, column q = l15
    const int jC = j0 + half;
    float partial = 0.f;
    const float* wbase = W + ((size_t)jC * UNITS + p) * (KLEN * QLEN) + l15;
#pragma unroll
    for (int r = 0; r < 8; ++r) partial += wbase[r * QLEN] * acc[r];
    // reduce over q (the 16 lanes of each half stay within their half for xor<=8)
    for (int m = 8; m; m >>= 1) partial += __shfl_xor(partial, m, 32);
    if (l15 == 0)
        b[(size_t)jC * UNITS + p] += partial;   // unique owner -> no atomics
}

// ------------------------------------------------------------------
extern "C" void kernel_launch(void* const* d_in, const int* in_sizes, int n_in,
                              void* d_out, int out_size, void* d_ws, size_t ws_size,
                              hipStream_t stream) {
    const float* x = (const float*)d_in[0];   // [128,2048,8]
    const float* W = (const float*)d_in[1];   // [2048,32,8,16]
    float* out = (float*)d_out;               // [128,32,16]
    float* ws  = (float*)d_ws;

    const size_t XT_N = (size_t)JCAPS * KLEN * B_SZ;   // 2,097,152
    const size_t BC_N = (size_t)JCAPS * UNITS;         // 65,536
    const size_t SV_N = (size_t)B_SZ * UNITS * QLEN;   // 65,536

    float* xt   = ws;
    float* bb   = xt + XT_N;
    float* cc   = bb + BC_N;
    float* ss   = cc + BC_N;
    float* vv   = ss + SV_N;
    float* pbuf = vv + SV_N;                           // CH * 65,536

    k_transpose_x<<<(int)(XT_N / 256), 256, 0, stream>>>(x, xt);
    k_zero<<<(int)((BC_N + 255) / 256), 256, 0, stream>>>(bb, (int)BC_N);

    for (int iter = 0; iter < 3; ++iter) {
        k_softmax<<<JCAPS / 8, 256, 0, stream>>>(bb, cc);
        k_s_gemm<<<UNITS * CH, 256, 0, stream>>>(x, W, cc, pbuf);
        k_reduce_s<<<(int)(SV_N / 256), 256, 0, stream>>>(pbuf, ss);
        float* vdst = (iter == 2) ? out : vv;
        k_squash<<<(B_SZ * UNITS) / 256, 256, 0, stream>>>(ss, vdst);
        if (iter < 2)
            k_b_update<<<(JCAPS / 2) * UNITS / 8, 256, 0, stream>>>(xt, vv, W, bb);
    }
}